// DAGBINNExactD1_55070070669887
// MI455X (gfx1250) — compile-verified
//
#include <hip/hip_runtime.h>
#include <hip/hip_bf16.h>

#define GENES 20000
#define LEVEL 10000
#define STEPS 7
#define DEG   16
#define NB    256          // batch
#define NC    2            // head outputs
#define EPS   (LEVEL * DEG)

typedef __attribute__((ext_vector_type(2))) float v2f;
typedef __attribute__((ext_vector_type(8))) float v8f;

// ---------------------------------------------------------------------------
// Tiled transpose: X[B][GENES] (row-major) -> Xt[GENES][B] (node-major,
// batch-minor) so that step-0 gathers are lane-coalesced.
// grid = (GENES/32, B/32), block = (32, 32)
// ---------------------------------------------------------------------------
__global__ void transpose_X_kernel(const float* __restrict__ X,
                                   float* __restrict__ Xt) {
    __shared__ float tile[32][33];  // +1 pad kills LDS bank conflicts
    const int gx = blockIdx.x * 32;   // gene base
    const int by = blockIdx.y * 32;   // batch base
    const int tx = threadIdx.x;
    const int ty = threadIdx.y;
    tile[ty][tx] = X[(size_t)(by + ty) * GENES + (gx + tx)];
    __syncthreads();
    Xt[(size_t)(gx + ty) * NB + (by + tx)] = tile[tx][ty];
}

// ---------------------------------------------------------------------------
// One propagation step. Layouts are node-major, batch-minor.
//   out[j][b] = tanh( bias[j] + sum_{d<16} in[src[j*16+d]-src_base][b] * w[j*16+d] )
// block = 256 threads (8 waves, lane == batch element), grid = LEVEL blocks.
// src/w addresses depend only on blockIdx -> scalarized (s_load) by compiler;
// the in[] gathers are 128B-coalesced per wave and L2-resident (10MB level).
// ---------------------------------------------------------------------------
__global__ void step_kernel(const float* __restrict__ in,    // [src_nodes][NB]
                            const float* __restrict__ w,     // EPS weights (this step)
                            const int*   __restrict__ src,   // EPS absolute src ids
                            const float* __restrict__ bias,  // LEVEL biases (this step)
                            float*       __restrict__ out,   // [LEVEL][NB]
                            int src_base) {
    const int j = blockIdx.x;      // destination node within level
    const int b = threadIdx.x;     // batch element
    float acc = 0.0f;
#pragma unroll
    for (int d = 0; d < DEG; ++d) {
        const int e = j * DEG + d;
        const int s = src[e] - src_base;          // block-uniform
        acc = fmaf(in[(size_t)s * NB + b], w[e], acc);
    }
    out[(size_t)j * NB + b] = tanhf(acc + bias[j]);
}

// ---------------------------------------------------------------------------
// Head GEMM via V_WMMA_F32_16X16X4_F32:
//   out[b][c] = sum_k roots[k][b] * W[k][c] + hb[c],  c < 2 (padded to N=16)
// One wave per 16-row batch tile; K=10000 consumed 8 at a time into two
// independent accumulators (breaks the WMMA D->C dependency chain).
// Operand packing per CDNA5 ISA 7.12.2:
//   A (16x4, 2 VGPR): lane L<16 -> M=L, K=v;   L>=16 -> M=L-16, K=v+2
//   B (4x16, 2 VGPR): lane L    -> N=L%16, K=v + 2*(L/16)
//   C/D (16x16, 8 VGPR): lane L -> N=L%16, M=v + 8*(L/16)
// B-fragment loads are unconditional (always-valid index kB*2+(n&1)) with a
// value-level select to zero -> v_cndmask, no EXEC-divergent branches.
// grid = NB/16 = 16 blocks, block = 32 threads (one wave32).
// ---------------------------------------------------------------------------
__global__ void head_wmma_kernel(const float* __restrict__ roots, // [LEVEL][NB]
                                 const float* __restrict__ W,     // [LEVEL][NC]
                                 const float* __restrict__ hb,    // [NC]
                                 float*       __restrict__ out) { // [NB][NC]
    const int lane  = threadIdx.x;        // 0..31
    const int mtile = blockIdx.x;         // batch tile: rows mtile*16 .. +15
    const int n     = lane & 15;          // column within 16-wide N
    const int half  = lane >> 4;          // upper/lower lane half
    const bool live = (n < NC);           // true for the 2 real head columns

    v8f acc0 = {};
    v8f acc1 = {};
    for (int kb = 0; kb < LEVEL; kb += 8) {
        v2f a0, b0, a1, b1;
#pragma unroll
        for (int v = 0; v < 2; ++v) {
            const int k0 = kb + v + 2 * half;        // K index per ISA layout
            const int k1 = k0 + 4;
            a0[v] = roots[(size_t)k0 * NB + (mtile * 16 + n)];
            a1[v] = roots[(size_t)k1 * NB + (mtile * 16 + n)];
            // Unconditional, always in-bounds load; mask via cndmask.
            const float w0 = W[(size_t)k0 * NC + (n & 1)];
            const float w1 = W[(size_t)k1 * NC + (n & 1)];
            b0[v] = live ? w0 : 0.0f;
            b1[v] = live ? w1 : 0.0f;
        }
        acc0 = __builtin_amdgcn_wmma_f32_16x16x4_f32(
            false, a0, false, b0, (short)0, acc0, false, false);
        acc1 = __builtin_amdgcn_wmma_f32_16x16x4_f32(
            false, a1, false, b1, (short)0, acc1, false, false);
    }

    if (live) {
        const float bias = hb[n];
#pragma unroll
        for (int v = 0; v < 8; ++v) {
            const int row = v + 8 * half;             // M of D element
            out[(size_t)(mtile * 16 + row) * NC + n] = acc0[v] + acc1[v] + bias;
        }
    }
}

// ---------------------------------------------------------------------------
// Inputs (setup_inputs order):
//  0:X (B*GENES f32)  1:edge_weight (E f32)  2:node_bias (N f32)
//  3:head_W (LEVEL*C) 4:head_b (C)           5:gene_map (identity, unused)
//  6:src (STEPS*EPS i32)  7:dst_pos (unused: dst = e/16)  8:dst_unique (unused:
//  contiguous)  9:eid (unused: contiguous)  10:root_ids (unused: last level)
// ---------------------------------------------------------------------------
extern "C" void kernel_launch(void* const* d_in, const int* in_sizes, int n_in,
                              void* d_out, int out_size, void* d_ws, size_t ws_size,
                              hipStream_t stream) {
    const float* X           = (const float*)d_in[0];
    const float* edge_weight = (const float*)d_in[1];
    const float* node_bias   = (const float*)d_in[2];
    const float* head_W      = (const float*)d_in[3];
    const float* head_b      = (const float*)d_in[4];
    const int*   src         = (const int*)  d_in[6];
    float*       out         = (float*)d_out;

    // Workspace layout (all f32): Xt | levA | levB  (~41 MB total)
    float* Xt   = (float*)d_ws;
    float* levA = Xt + (size_t)GENES * NB;
    float* levB = levA + (size_t)LEVEL * NB;

    // 1) Transpose gene activations to node-major.
    dim3 tb(32, 32);
    dim3 tg(GENES / 32, NB / 32);
    transpose_X_kernel<<<tg, tb, 0, stream>>>(X, Xt);

    // 2) Seven propagation steps, ping-ponging level buffers.
    for (int k = 0; k < STEPS; ++k) {
        const float* in  = (k == 0) ? Xt : (((k - 1) % 2 == 0) ? levA : levB);
        float*       dst = (k % 2 == 0) ? levA : levB;
        const int src_base = (k == 0) ? 0 : (GENES + (k - 1) * LEVEL);
        step_kernel<<<LEVEL, NB, 0, stream>>>(
            in,
            edge_weight + (size_t)k * EPS,      // eid[k] is contiguous
            src + (size_t)k * EPS,
            node_bias + GENES + (size_t)k * LEVEL,  // dst_unique[k] contiguous
            dst, src_base);
    }

    // 3) Dense head via f32 WMMA. roots = output of step 6 -> levA.
    head_wmma_kernel<<<NB / 16, 32, 0, stream>>>(levA, head_W, head_b, out);
}